// VoxelFeatureRegionAggregation_5317169512796
// MI455X (gfx1250) — compile-verified
//
#include <hip/hip_runtime.h>
#include <hip/hip_bf16.h>

// Problem geometry (matches reference setup_inputs; window scalars live on
// device so grid shapes use these compile-time constants).
#define WXN 128
#define WYN 128
#define NBATCH 4
#define NW (NBATCH * WYN * WXN)   // 65536 dense window slots
#define FD 256                    // feature dim
#define POS_TEMP 10000.0f

typedef float v2f __attribute__((ext_vector_type(2)));
typedef float v8f __attribute__((ext_vector_type(8)));

// ---------------------------------------------------------------------------
// K0: zero the dense feature + pos-emb output regions and the counts buffer.
// (harness poisons d_out once; we must re-init every call)
// ---------------------------------------------------------------------------
__global__ void vfra_zero_kernel(float* __restrict__ out, float* __restrict__ counts) {
  size_t i = (size_t)blockIdx.x * blockDim.x + threadIdx.x;
  const size_t nf4 = (size_t)2 * NW * FD / 4;  // feat + pos regions, float4 units
  if (i < nf4) ((float4*)out)[i] = make_float4(0.f, 0.f, 0.f, 0.f);
  if (i < NW) counts[i] = 0.f;
}

// ---------------------------------------------------------------------------
// K1: scatter-accumulate voxel features into the dense (NW, FD) buffer.
// One 256-thread block (8 wave32) handles 8 voxels; each thread owns one
// feature column -> fully coalesced 1KB rows. Accumulator (67MB) lives in L2,
// so the f32 atomics resolve on-chip. Prefetch the next row while adding.
// ---------------------------------------------------------------------------
__global__ void vfra_scatter_kernel(const float* __restrict__ feat,
                                    const int* __restrict__ winds,
                                    float* __restrict__ acc,
                                    float* __restrict__ counts,
                                    int M) {
  const int VPB = 8;
  int v0 = blockIdx.x * VPB;
  int tid = threadIdx.x;  // 0..FD-1
  for (int i = 0; i < VPB; ++i) {
    int v = v0 + i;
    if (v >= M) break;
    int b = winds[v * 4 + 0];
    int z = winds[v * 4 + 1];
    int y = winds[v * 4 + 2];
    int x = winds[v * 4 + 3];
    int key = ((b * 1 + z) * WYN + y) * WXN + x;  // Z_CAP == 1
    if (v + 1 < M)  // gfx1250: global_prefetch_b8
      __builtin_prefetch(&feat[(size_t)(v + 1) * FD + tid], 0, 0);
    if (tid == 0) atomicAdd(&counts[key], 1.0f);
    atomicAdd(&acc[(size_t)key * FD + tid], feat[(size_t)v * FD + tid]);
  }
}

// ---------------------------------------------------------------------------
// K2: in-place row scaling sums -> means via the matrix pipe.
// For each 16-window x 16-feature tile F, compute D = diag(r) * F with four
// chained V_WMMA_F32_16X16X4_F32 ops (K-slices of the diagonal).
// Wave32 layouts per ISA 7.12.2:
//   A (16x4 f32): lane l holds M=l&15; comp0 = K(0|2 by lane half), comp1 = K+1
//   B (4x16 f32): lane l holds N=l&15; comp0 = K(0|2 by lane half), comp1 = K+1
//   C/D (16x16 f32): lane l holds N=l&15; VGPR v = row M = v + 8*(l>>4)
// Block = 256 threads = 8 waves; each wave walks feature chunks.
// ---------------------------------------------------------------------------
__global__ void vfra_scale_wmma_kernel(float* __restrict__ acc,
                                       const float* __restrict__ counts) {
  int w0 = blockIdx.x * 16;           // 16-window tile base
  int lane = threadIdx.x & 31;
  int wave = threadIdx.x >> 5;        // 0..7
  int m = lane & 15;
  int half = lane >> 4;               // 0: lanes 0-15, 1: lanes 16-31
  int k0 = half * 2;                  // K index of vector component 0

  float cnt = counts[w0 + m];
  float r = 1.0f / fmaxf(cnt, 1.0f);  // reference: sums / max(counts, 1)

  for (int cc = wave; cc < FD / 16; cc += 8) {
    int c0 = cc * 16;
    v8f d = {};
#pragma unroll
    for (int j = 0; j < 4; ++j) {
      v2f a, b;
      // A_j = columns [4j..4j+3] of diag(r): A(m,k) = (m == 4j+k) ? r_m : 0
      a.x = (m == 4 * j + k0)     ? r : 0.0f;
      a.y = (m == 4 * j + k0 + 1) ? r : 0.0f;
      // B_j = rows [4j..4j+3] of the accumulated tile F
      b.x = acc[(size_t)(w0 + 4 * j + k0)     * FD + c0 + m];
      b.y = acc[(size_t)(w0 + 4 * j + k0 + 1) * FD + c0 + m];
      d = __builtin_amdgcn_wmma_f32_16x16x4_f32(
          /*neg_a=*/false, a, /*neg_b=*/false, b,
          /*c_mod=*/(short)0, d, /*reuse_a=*/false, /*reuse_b=*/false);
    }
#pragma unroll
    for (int v = 0; v < 8; ++v)
      acc[(size_t)(w0 + v + 8 * half) * FD + c0 + m] = d[v];
  }
}

// ---------------------------------------------------------------------------
// K3: positional embedding + key-padding mask for every dense window slot.
// out[p] = (p even ? sin : cos)(c / POS_TEMP^((p>>1)/64)); x-half then y-half.
// Unoccupied slots stay 0 (reference scatters into zeros); padding = !occ.
// ---------------------------------------------------------------------------
__global__ void vfra_posemb_kernel(const float* __restrict__ counts,
                                   float* __restrict__ pos,
                                   float* __restrict__ pad) {
  int w = blockIdx.x;       // dense window id
  int d = threadIdx.x;      // 0..FD-1
  float occ = counts[w] > 0.0f ? 1.0f : 0.0f;
  int xw = w & (WXN - 1);
  int yw = (w >> 7) & (WYN - 1);
  float c = (d < FD / 2) ? ((float)xw - WXN * 0.5f)
                         : ((float)yw - WYN * 0.5f);
  int p = (d < FD / 2) ? d : d - FD / 2;
  float invf = __powf(POS_TEMP, (float)(p >> 1) / 64.0f);
  float e = c / invf;
  float val = (p & 1) ? __cosf(e) : __sinf(e);
  pos[(size_t)w * FD + d] = occ * val;
  if (d == 0) pad[w] = 1.0f - occ;
}

// ---------------------------------------------------------------------------
// K4: sorted-unique outputs. Sorted unique keys == occupied dense slots in
// increasing order, so rank = exclusive prefix-sum of occupancy. One 1024-
// thread block scans the 65536 slots in 64 LDS passes (counts < 2^24, exact
// in f32) and emits flat2batch_inds + decoded (b,z,y,x) coords.
// ---------------------------------------------------------------------------
__global__ void vfra_rank_kernel(const float* __restrict__ counts,
                                 float* __restrict__ out_inds,    // (Mp,)
                                 float* __restrict__ out_coord,   // (Mp,4)
                                 int Mp) {
  __shared__ float s[1024];
  __shared__ float base;
  int tid = threadIdx.x;
  if (tid == 0) base = 0.f;
  __syncthreads();
  for (int start = 0; start < NW; start += 1024) {
    int w = start + tid;
    float occ = counts[w] > 0.f ? 1.f : 0.f;
    s[tid] = occ;
    __syncthreads();
    for (int off = 1; off < 1024; off <<= 1) {   // inclusive Hillis-Steele
      float t = (tid >= off) ? s[tid - off] : 0.f;
      __syncthreads();
      s[tid] += t;
      __syncthreads();
    }
    if (occ > 0.f) {
      int rank = (int)(base + s[tid]) - 1;
      if (rank >= 0 && rank < Mp) {
        out_inds[rank] = (float)w;
        int x = w & (WXN - 1);
        int y = (w >> 7) & (WYN - 1);
        int b = w >> 14;
        out_coord[rank * 4 + 0] = (float)b;
        out_coord[rank * 4 + 1] = 0.f;           // z
        out_coord[rank * 4 + 2] = (float)y;
        out_coord[rank * 4 + 3] = (float)x;
      }
    }
    __syncthreads();
    if (tid == 1023) base += s[1023];
    __syncthreads();
  }
}

// ---------------------------------------------------------------------------
extern "C" void kernel_launch(void* const* d_in, const int* in_sizes, int n_in,
                              void* d_out, int out_size, void* d_ws, size_t ws_size,
                              hipStream_t stream) {
  const float* feat = (const float*)d_in[0];
  const int* winds  = (const int*)d_in[1];
  float* out = (float*)d_out;

  int M = in_sizes[1] / 4;           // (M,4) window indices

  // Output layout (concatenated flat, return order):
  //   [0]              voxel_feat_win_batch : NW*FD
  //   [O1]             pos_emb_win_batch    : NW*FD
  //   [O2]             key_padding          : NW
  //   [O3]             flat2batch_inds      : Mp
  //   [O3+Mp]          voxel_coord_win      : Mp*4
  const size_t O1 = (size_t)NW * FD;
  const size_t O2 = 2 * O1;
  const size_t O3 = O2 + NW;
  int Mp = (int)(((size_t)out_size - O3) / 5);

  float* out_feat  = out;
  float* out_pos   = out + O1;
  float* out_pad   = out + O2;
  float* out_inds  = out + O3;
  float* out_coord = out + O3 + (size_t)Mp;

  float* counts = (float*)d_ws;      // NW floats of scratch

  // K0: zero feat/pos regions + counts
  {
    size_t nf4 = (size_t)2 * NW * FD / 4;
    int blocks = (int)((nf4 + 255) / 256);
    vfra_zero_kernel<<<blocks, 256, 0, stream>>>(out, counts);
  }
  // K1: dense scatter-accumulate
  vfra_scatter_kernel<<<(M + 7) / 8, 256, 0, stream>>>(feat, winds, out_feat,
                                                       counts, M);
  // K2: sums -> means via WMMA diag-scale (16-window tiles)
  vfra_scale_wmma_kernel<<<NW / 16, 256, 0, stream>>>(out_feat, counts);
  // K3: positional embedding + padding mask
  vfra_posemb_kernel<<<NW, FD, 0, stream>>>(counts, out_pos, out_pad);
  // K4: sorted-unique ids + coords via occupancy scan
  vfra_rank_kernel<<<1, 1024, 0, stream>>>(counts, out_inds, out_coord, Mp);
}